// RGCNModel_73289321939191
// MI455X (gfx1250) — compile-verified
//
#include <hip/hip_runtime.h>

#define Nn 50000
#define Ee 800000
#define Dd 128
#define Rr 8
#define Bb 4
#define Ll 2
#define BN_EPS 1e-3f

typedef __attribute__((ext_vector_type(2))) float v2f;
typedef __attribute__((ext_vector_type(8))) float v8f;

// ---------------------------------------------------------------- utilities
__global__ void zero_f32(float* __restrict__ p, int n) {
  int i = blockIdx.x * blockDim.x + threadIdx.x;
  if (i < n) p[i] = 0.0f;
}

// per-(relation, dst) incoming-edge counts
__global__ void count_kernel(const int* __restrict__ etype,
                             const int* __restrict__ dst,
                             int* __restrict__ cnt) {
  int e = blockIdx.x * blockDim.x + threadIdx.x;
  if (e < Ee) atomicAdd(&cnt[etype[e] * Nn + dst[e]], 1);
}

// node_norm[v] = 1/cnt[r][v] for the LAST r (ascending) with cnt>0, else 0
__global__ void norm_kernel(const int* __restrict__ cnt,
                            float* __restrict__ norm) {
  int v = blockIdx.x * blockDim.x + threadIdx.x;
  if (v >= Nn) return;
  float nv = 0.0f;
#pragma unroll
  for (int r = 0; r < Rr; ++r) {
    int c = cnt[r * Nn + v];
    if (c > 0) nv = 1.0f / (float)c;
  }
  norm[v] = nv;
}

// W[r] = sum_b w_coe[r,b] * bases[b]    (R*D*D elements)
__global__ void coeff_kernel(const float* __restrict__ w_coe,
                             const float* __restrict__ bases,
                             float* __restrict__ W) {
  int i = blockIdx.x * blockDim.x + threadIdx.x;
  if (i >= Rr * Dd * Dd) return;
  int r = i / (Dd * Dd);
  int rem = i - r * (Dd * Dd);
  float acc = 0.0f;
#pragma unroll
  for (int b = 0; b < Bb; ++b)
    acc += w_coe[r * Bb + b] * bases[b * Dd * Dd + rem];
  W[i] = acc;
}

// ------------------------------------------------------------- WMMA GEMM
// C[N x 128] = A[N x 128] @ Bm[128 x 128], fused epilogues:
//   mode 0: plain store
//   mode 1: + bias[col]
//   mode 2: v += agg[row,col]; BN(gamma,beta,mean,var); ReLU   (combine)
// One wave -> 16 rows x 128 cols (8 accumulators of v8f).
// Fragment layouts per CDNA5 ISA 7.12.2 (f32 16x16x4).
// A rows >= Nn are clamped to row Nn-1 instead of predicated: an A row M only
// feeds output row M, and out-of-range output rows are never stored, so the
// clamp is branch-free-safe and lets the two A elements merge into one b64.
__global__ __launch_bounds__(256) void gemm128(
    const float* __restrict__ A, const float* __restrict__ Bm,
    float* __restrict__ C, int mode,
    const float* __restrict__ bias, const float* __restrict__ agg,
    const float* __restrict__ gamma, const float* __restrict__ beta,
    const float* __restrict__ mean, const float* __restrict__ var) {
  const int wave = threadIdx.x >> 5;
  const int lane = threadIdx.x & 31;
  const int half = lane >> 4;   // 0: lanes 0-15, 1: lanes 16-31
  const int l16  = lane & 15;
  const long row0 = (long)blockIdx.x * 128 + wave * 16;

  long arow_i = row0 + l16;                 // A row for this lane (both halves)
  if (arow_i >= Nn) arow_i = Nn - 1;        // clamp (see note above)
  const float* arow = A + arow_i * Dd + 2 * half;

  v8f acc[8] = {};

#pragma unroll 4
  for (int k0 = 0; k0 < Dd; k0 += 4) {
    // A 16x4 fragment: v0 = K(+0|+2), v1 = K(+1|+3) by lane half -> one b64
    v2f a = *(const v2f*)(arow + k0);
#pragma unroll
    for (int nt = 0; nt < 8; ++nt) {
      const float* bp = Bm + (k0 + 2 * half) * Dd + nt * 16 + l16;
      v2f b;
      b.x = bp[0];      // row K,   col nt*16+l16  (64B-coalesced across lanes)
      b.y = bp[Dd];     // row K+1, col nt*16+l16
      acc[nt] = __builtin_amdgcn_wmma_f32_16x16x4_f32(
          false, a, false, b, (short)0, acc[nt], false, false);
    }
  }

  // Epilogue: D VGPR i -> row (row0 + i + 8*half), col nt*16 + l16
#pragma unroll
  for (int nt = 0; nt < 8; ++nt) {
    const int col = nt * 16 + l16;
    float g = 1.0f, bt = 0.0f, mu = 0.0f, iv = 1.0f, bs = 0.0f;
    if (mode == 2) {
      g  = gamma[col];
      bt = beta[col];
      mu = mean[col];
      iv = rsqrtf(var[col] + BN_EPS);
    } else if (mode == 1) {
      bs = bias[col];
    }
#pragma unroll
    for (int i = 0; i < 8; ++i) {
      long r = row0 + 8 * half + i;
      if (r < Nn) {
        float v = acc[nt][i];
        if (mode == 1) {
          v += bs;
        } else if (mode == 2) {
          v += agg[r * Dd + col];
          v = g * (v - mu) * iv + bt;
          v = v > 0.0f ? v : 0.0f;
        }
        C[r * Dd + col] = v;
      }
    }
  }
}

// --------------------------------------------------- per-relation scatter
// For edges of type r: agg[dst] += norm[dst] * T[src]   (one wave per edge)
__global__ __launch_bounds__(256) void scatter_kernel(
    const int* __restrict__ src, const int* __restrict__ dst,
    const int* __restrict__ etype, const float* __restrict__ node_norm,
    const float* __restrict__ T, float* __restrict__ agg, int r) {
  int e = blockIdx.x * 8 + (threadIdx.x >> 5);
  int lane = threadIdx.x & 31;
  if (e >= Ee) return;
  if (etype[e] != r) return;
  int s = src[e], d = dst[e];
  float nrm = node_norm[d];
  const float4* tp = (const float4*)(T + (long)s * Dd);
  float4 v = tp[lane];                       // 32 lanes x float4 = 128
  float* ap = agg + (long)d * Dd + lane * 4;
  atomicAdd(ap + 0, nrm * v.x);
  atomicAdd(ap + 1, nrm * v.y);
  atomicAdd(ap + 2, nrm * v.z);
  atomicAdd(ap + 3, nrm * v.w);
}

// ------------------------------------------------------------------ launch
extern "C" void kernel_launch(void* const* d_in, const int* in_sizes, int n_in,
                              void* d_out, int out_size, void* d_ws,
                              size_t ws_size, hipStream_t stream) {
  (void)in_sizes; (void)n_in; (void)out_size; (void)ws_size;
  const float* node_feat = (const float*)d_in[0];
  const float* W_emb     = (const float*)d_in[1];
  const float* b_emb     = (const float*)d_in[2];
  const float* bases     = (const float*)d_in[3];
  const float* w_coe     = (const float*)d_in[4];
  const float* self_loop = (const float*)d_in[5];
  const float* bn_gamma  = (const float*)d_in[6];
  const float* bn_beta   = (const float*)d_in[7];
  const float* bn_mean   = (const float*)d_in[8];
  const float* bn_var    = (const float*)d_in[9];
  const int*   src       = (const int*)d_in[10];
  const int*   dst       = (const int*)d_in[11];
  const int*   etype     = (const int*)d_in[12];

  // workspace carve-up (256B aligned)
  char* ws = (char*)d_ws;
  size_t o = 0;
  auto take = [&](size_t bytes) {
    void* p = ws + o;
    o += (bytes + 255) & ~(size_t)255;
    return p;
  };
  float* h    = (float*)take((size_t)Nn * Dd * sizeof(float));   // 25.6 MB
  float* T    = (float*)take((size_t)Nn * Dd * sizeof(float));   // 25.6 MB
  float* agg  = (float*)take((size_t)Nn * Dd * sizeof(float));   // 25.6 MB
  float* Wb   = (float*)take((size_t)Rr * Dd * Dd * sizeof(float));
  float* nrm  = (float*)take((size_t)Nn * sizeof(float));
  int*   cnt  = (int*)take((size_t)Rr * Nn * sizeof(int));

  const int TB = 256;
  const int gemm_blocks = (Nn + 127) / 128;

  // --- edge normalization (overwrite semantics across relations) ---
  zero_f32<<<(Rr * Nn + TB - 1) / TB, TB, 0, stream>>>((float*)cnt, Rr * Nn);
  count_kernel<<<(Ee + TB - 1) / TB, TB, 0, stream>>>(etype, dst, cnt);
  norm_kernel<<<(Nn + TB - 1) / TB, TB, 0, stream>>>(cnt, nrm);

  // --- embedding: h = node_feat @ W_emb + b_emb ---
  gemm128<<<gemm_blocks, TB, 0, stream>>>(node_feat, W_emb, h, 1, b_emb,
                                          nullptr, nullptr, nullptr, nullptr,
                                          nullptr);

  for (int l = 0; l < Ll; ++l) {
    // basis decomposition -> Wb[r]
    coeff_kernel<<<(Rr * Dd * Dd + TB - 1) / TB, TB, 0, stream>>>(
        w_coe + (size_t)l * Rr * Bb, bases + (size_t)l * Bb * Dd * Dd, Wb);
    zero_f32<<<(Nn * Dd + TB - 1) / TB, TB, 0, stream>>>(agg, Nn * Dd);

    for (int r = 0; r < Rr; ++r) {
      // T = h @ W[r]
      gemm128<<<gemm_blocks, TB, 0, stream>>>(h, Wb + (size_t)r * Dd * Dd, T, 0,
                                              nullptr, nullptr, nullptr,
                                              nullptr, nullptr, nullptr);
      // agg[dst] += norm[dst] * T[src] over edges of type r
      scatter_kernel<<<(Ee + 7) / 8, TB, 0, stream>>>(src, dst, etype, nrm, T,
                                                      agg, r);
    }

    // h_next = relu(BN(agg + h @ self_loop[l]))
    float* outp = (l == Ll - 1) ? (float*)d_out : h;
    gemm128<<<gemm_blocks, TB, 0, stream>>>(
        h, self_loop + (size_t)l * Dd * Dd, outp, 2, nullptr, agg,
        bn_gamma + (size_t)l * Dd, bn_beta + (size_t)l * Dd,
        bn_mean + (size_t)l * Dd, bn_var + (size_t)l * Dd);
  }
}